// VariableSelectionNetwork_25400436588741
// MI455X (gfx1250) — compile-verified
//
#include <hip/hip_runtime.h>
#include <hip/hip_bf16.h>
#include <math.h>

// MI455X / gfx1250 fused VariableSelectionNetwork, fp32 WMMA path.
// B=16384, V=16, S=64, H=512.
// v3: W pre-interleaved to [v][s/2][h][2] so each B-operand pair is one
// global_load_b64 -> inner loop is 4 VMEM per 4 WMMAs (was 6).

typedef __attribute__((ext_vector_type(2))) float v2f;
typedef __attribute__((ext_vector_type(8))) float v8f;

#define BDIM 16384
#define VDIM 16
#define SDIM 64
#define HDIM 512
#define ROWS 32  // batch rows per workgroup (two 16-row M-tiles)

static __device__ inline v8f wmma_f32_16x16x4(v2f a, v2f b, v8f c) {
  // (neg_a, A, neg_b, B, c_mod, C, reuse_a, reuse_b)
  return __builtin_amdgcn_wmma_f32_16x16x4_f32(false, a, false, b, (short)0, c,
                                               false, false);
}

// ---------------------------------------------------------------------------
// Prep: W[v][s][h] -> W2[v][s/2][h][2] so the two K-values a lane feeds to a
// WMMA B operand (s, s+1 at fixed h) are adjacent -> one b64 load per WMMA.
// 512K elements, coalesced read; runs once per call, stays hot in L2.
// ---------------------------------------------------------------------------
__global__ __launch_bounds__(256) void vsn_prep(const float* __restrict__ W,
                                                float* __restrict__ W2) {
  const int idx = blockIdx.x * 256 + threadIdx.x;  // [v][s][h] linear
  const int h = idx & (HDIM - 1);
  const int s = (idx >> 9) & (SDIM - 1);
  const int v = idx >> 15;
  W2[(size_t)v * (SDIM * HDIM) + (((s >> 1) * HDIM + h) << 1) + (s & 1)] =
      W[idx];
}

// ---------------------------------------------------------------------------
// Phase 1: flat = sum_v (x@W[v] + b[v]) kept in LDS, then
// logits = flat/16 @ W_sel + b_sel (waves 0..1, WMMA over K=512),
// then per-row softmax -> weights[B,V] written to workspace.
// One workgroup = 32 batch rows; 16 waves, wave w owns H-cols [32w, 32w+32).
// ---------------------------------------------------------------------------
__global__ __launch_bounds__(512) void vsn_phase1(
    const float* __restrict__ x, const float* __restrict__ W2,
    const float* __restrict__ bias, const float* __restrict__ Wsel,
    const float* __restrict__ bsel, float* __restrict__ wts) {
  __shared__ float flat[ROWS * HDIM];  // 64 KB
  __shared__ float lg[ROWS * VDIM];    // 2 KB logits tile

  const int tid = threadIdx.x;
  const int wave = tid >> 5;
  const int lane = tid & 31;
  const int m = lane & 15;   // A row / B,C column
  const int hi = lane >> 4;  // lane-half (K/M offset select)
  const int b0 = blockIdx.x * ROWS;

  const int h0 = wave * 32 + m;  // first owned N-tile column
  const int h1 = h0 + 16;        // second owned N-tile column

  v8f f00 = {}, f01 = {};  // rows b0..b0+15   x  {h0-tile, h1-tile}
  v8f f10 = {}, f11 = {};  // rows b0+16..+31  x  {h0-tile, h1-tile}

  for (int v = 0; v < VDIM; ++v) {
    const float* xp0 =
        x + (size_t)(b0 + m) * (VDIM * SDIM) + (size_t)v * SDIM + 2 * hi;
    const float* xp1 = xp0 + (size_t)16 * (VDIM * SDIM);
    // B pair for k-step k: W2 at s2 = 2k + hi, column h -> contiguous float2.
    const float* w2v = W2 + (size_t)v * (SDIM * HDIM) +
                       (size_t)hi * (2 * HDIM) + (size_t)2 * h0;
#pragma unroll
    for (int k = 0; k < 16; ++k) {
      v2f a0 = *(const v2f*)(xp0 + 4 * k);  // A rows 0..15
      v2f a1 = *(const v2f*)(xp1 + 4 * k);  // A rows 16..31
      const float* wk = w2v + (size_t)k * (4 * HDIM);
      v2f bb0 = *(const v2f*)(wk);       // cols h0-tile
      v2f bb1 = *(const v2f*)(wk + 32);  // cols h1-tile (+16 cols * 2)
      // each B pair feeds two M-tiles
      f00 = wmma_f32_16x16x4(a0, bb0, f00);
      f01 = wmma_f32_16x16x4(a0, bb1, f01);
      f10 = wmma_f32_16x16x4(a1, bb0, f10);
      f11 = wmma_f32_16x16x4(a1, bb1, f11);
    }
  }

  // Column sums of bias over v (flat holds sum over v of transformed).
  float bs0 = 0.0f, bs1 = 0.0f;
  for (int v = 0; v < VDIM; ++v) {
    bs0 += bias[v * HDIM + h0];
    bs1 += bias[v * HDIM + h1];
  }

#pragma unroll
  for (int r = 0; r < 8; ++r) {
    const int row = r + 8 * hi;  // C/D layout: lanes 16-31 hold rows 8..15
    flat[row * HDIM + h0] = f00[r] + bs0;
    flat[row * HDIM + h1] = f01[r] + bs1;
    flat[(row + 16) * HDIM + h0] = f10[r] + bs0;
    flat[(row + 16) * HDIM + h1] = f11[r] + bs1;
  }
  __syncthreads();

  // Selection GEMM: logits[32,16] = (flat/16) @ W_sel + b_sel, two waves.
  if (wave < 2) {
    const int rb = wave * 16;  // this wave's 16-row slice
    v8f lacc = {};
    for (int kk = 0; kk < HDIM / 4; ++kk) {
      const int s = 4 * kk + 2 * hi;
      v2f a = *(const v2f*)&flat[(rb + m) * HDIM + s];  // 8B-aligned LDS read
      v2f bb;
      bb.x = Wsel[(size_t)s * VDIM + m];
      bb.y = Wsel[(size_t)(s + 1) * VDIM + m];
      lacc = wmma_f32_16x16x4(a, bb, lacc);
    }
    const float bv = bsel[m];
#pragma unroll
    for (int r = 0; r < 8; ++r) {
      const int row = rb + r + 8 * hi;
      lg[row * VDIM + m] = lacc[r] * (1.0f / VDIM) + bv;  // fold mean(1/16)
    }
  }
  __syncthreads();

  // Row softmax over V=16 and store weights to workspace.
  if (tid < ROWS) {
    float e[VDIM];
    float mx = -INFINITY;
    for (int j = 0; j < VDIM; ++j) mx = fmaxf(mx, lg[tid * VDIM + j]);
    float sum = 0.0f;
    for (int j = 0; j < VDIM; ++j) {
      e[j] = __expf(lg[tid * VDIM + j] - mx);
      sum += e[j];
    }
    const float inv = 1.0f / sum;
    for (int j = 0; j < VDIM; ++j)
      wts[(size_t)(b0 + tid) * VDIM + j] = e[j] * inv;
  }
}

// ---------------------------------------------------------------------------
// Phase 2: recompute transformed tiles (W2 stays hot in 192MB L2) and fold the
// softmax weights: out[b,h] = sum_v (x@W[v] + b[v]) * wt[b,v]. Each wave owns
// disjoint columns -> pure register accumulation, direct global store.
// ---------------------------------------------------------------------------
__global__ __launch_bounds__(512) void vsn_phase2(
    const float* __restrict__ x, const float* __restrict__ W2,
    const float* __restrict__ bias, const float* __restrict__ wts,
    float* __restrict__ out) {
  __shared__ float wt[ROWS * VDIM];  // weights tile for these 32 rows

  const int tid = threadIdx.x;
  const int wave = tid >> 5;
  const int lane = tid & 31;
  const int m = lane & 15;
  const int hi = lane >> 4;
  const int b0 = blockIdx.x * ROWS;

  wt[tid] = wts[(size_t)b0 * VDIM + tid];  // 512 floats, one per thread
  __syncthreads();

  const int h0 = wave * 32 + m;
  const int h1 = h0 + 16;

  v8f o00 = {}, o01 = {};  // weighted-sum accumulators, rows 0..15
  v8f o10 = {}, o11 = {};  // rows 16..31

  for (int v = 0; v < VDIM; ++v) {
    const float* xp0 =
        x + (size_t)(b0 + m) * (VDIM * SDIM) + (size_t)v * SDIM + 2 * hi;
    const float* xp1 = xp0 + (size_t)16 * (VDIM * SDIM);
    const float* w2v = W2 + (size_t)v * (SDIM * HDIM) +
                       (size_t)hi * (2 * HDIM) + (size_t)2 * h0;

    v8f a00 = {}, a01 = {}, a10 = {}, a11 = {};
#pragma unroll
    for (int k = 0; k < 16; ++k) {
      v2f a0 = *(const v2f*)(xp0 + 4 * k);
      v2f a1 = *(const v2f*)(xp1 + 4 * k);
      const float* wk = w2v + (size_t)k * (4 * HDIM);
      v2f bb0 = *(const v2f*)(wk);
      v2f bb1 = *(const v2f*)(wk + 32);
      a00 = wmma_f32_16x16x4(a0, bb0, a00);
      a01 = wmma_f32_16x16x4(a0, bb1, a01);
      a10 = wmma_f32_16x16x4(a1, bb0, a10);
      a11 = wmma_f32_16x16x4(a1, bb1, a11);
    }
    const float bb0s = bias[v * HDIM + h0];
    const float bb1s = bias[v * HDIM + h1];
#pragma unroll
    for (int r = 0; r < 8; ++r) {
      const int row = r + 8 * hi;
      const float wv0 = wt[row * VDIM + v];
      const float wv1 = wt[(row + 16) * VDIM + v];
      o00[r] += (a00[r] + bb0s) * wv0;
      o01[r] += (a01[r] + bb1s) * wv0;
      o10[r] += (a10[r] + bb0s) * wv1;
      o11[r] += (a11[r] + bb1s) * wv1;
    }
  }

#pragma unroll
  for (int r = 0; r < 8; ++r) {
    const int row = b0 + r + 8 * hi;
    out[(size_t)row * HDIM + h0] = o00[r];
    out[(size_t)row * HDIM + h1] = o01[r];
    out[(size_t)(row + 16) * HDIM + h0] = o10[r];
    out[(size_t)(row + 16) * HDIM + h1] = o11[r];
  }
}

extern "C" void kernel_launch(void* const* d_in, const int* in_sizes, int n_in,
                              void* d_out, int out_size, void* d_ws,
                              size_t ws_size, hipStream_t stream) {
  const float* x = (const float*)d_in[0];     // [B,V,S]
  const float* W = (const float*)d_in[1];     // [V,S,H]
  const float* b = (const float*)d_in[2];     // [V,H]
  const float* Wsel = (const float*)d_in[3];  // [H,V]
  const float* bsel = (const float*)d_in[4];  // [V]
  float* out = (float*)d_out;                 // [B,H]

  float* wts = (float*)d_ws;  // [B,V] softmax weights (1 MB)
  float* W2 = (float*)((char*)d_ws + (size_t)BDIM * VDIM * sizeof(float));
  // W2: pair-interleaved W, 2 MB, recomputed every call (deterministic).

  vsn_prep<<<dim3((VDIM * SDIM * HDIM) / 256), 256, 0, stream>>>(W, W2);
  dim3 grid(BDIM / ROWS);
  vsn_phase1<<<grid, 512, 0, stream>>>(x, W2, b, Wsel, bsel, wts);
  vsn_phase2<<<grid, 512, 0, stream>>>(x, W2, b, wts, out);
}